// Semantic_Mapping_71949292142944
// MI455X (gfx1250) — compile-verified
//
#include <hip/hip_runtime.h>
#include <math.h>
#include <stdint.h>

// ---------------- problem constants (match reference) ----------------
#define H_IMG   1024
#define W_IMG   1280
#define HW      (H_IMG * W_IMG)        // 1310720
#define DEPTH_OFF (3 * HW)             // obs channel 3 = depth
#define VR      100                    // voxel grid x,y
#define HZ      88                     // voxel grid z (72 - (-16))
#define MINZ    21                     // int(25/5) + 16
#define MAXZ    33                     // int(89/5) + 16
#define MAPC    480                    // MAP_CELLS
#define MC2     (MAPC * MAPC)          // 230400
#define WIN_Y0  240                    // agent-view window rows [240,340)
#define WIN_X0  190                    // agent-view window cols [190,290)
#define DEGF    57.29577951308232f
#define RAD_PER_DEG 0.017453292519943295f

// splat launch geometry (exact tiling: 128*256*4 px per sweep, 10 sweeps)
#define SPLAT_BLOCKS  128
#define SPLAT_THREADS 256
#define PIX_PER_THREAD 4
#define CHUNK_PIX     (SPLAT_THREADS * PIX_PER_THREAD)        // 1024
#define CHUNK_STRIDE  (SPLAT_BLOCKS * CHUNK_PIX)              // 131072
#define N_CHUNKS      (HW / CHUNK_STRIDE)                     // 10 (exact)

// d_out layout (floats): fp_map_pred[10000] | map_pred[4*230400] | pose[3] | pose[3]
#define OUT_FP   0
#define OUT_MAP  (VR * VR)
#define OUT_POSE (VR * VR + 4 * MC2)

// d_ws layout (floats): map_sum[10000] | exp_sum[10000] | rotated[2*230400]
#define WS_MAP 0
#define WS_EXP (VR * VR)
#define WS_ROT (2 * VR * VR)

typedef float v4f __attribute__((ext_vector_type(4)));

// ---- CDNA5 async global->LDS copy (ASYNCcnt-tracked), inline asm for
// toolchain portability (bypasses the 5-arg vs 6-arg builtin mismatch) ----
__device__ __forceinline__ void async_load_b128(uint32_t lds_byte_addr,
                                                const void* gaddr) {
  uint64_t g = (uint64_t)(uintptr_t)gaddr;
  asm volatile("global_load_async_to_lds_b128 %0, %1, off"
               :: "v"(lds_byte_addr), "v"(g) : "memory");
}
__device__ __forceinline__ void wait_async_le1() {
  asm volatile("s_wait_asynccnt 0x1" ::: "memory");
}
__device__ __forceinline__ void wait_async_0() {
  asm volatile("s_wait_asynccnt 0x0" ::: "memory");
}

// ---------------- pose math (shared by several kernels) ----------------
__device__ __forceinline__ void eval_pose(const float* __restrict__ pose_obs,
                                          const float* __restrict__ poses_last,
                                          float& px, float& py, float& nt) {
  float th = poses_last[2] * (1.0f / DEGF);
  float s = sinf(th), c = cosf(th);
  float dx = pose_obs[0] * c - pose_obs[1] * s;
  float dy = pose_obs[0] * s + pose_obs[1] * c;
  float t = poses_last[2] + pose_obs[2] * DEGF;
  t = fmodf(t - 180.0f, 360.0f) + 180.0f;
  t = fmodf(t + 180.0f, 360.0f) - 180.0f;
  px = poses_last[0] + dx;
  py = poses_last[1] + dy;
  nt = t;
}

__device__ __forceinline__ void eval_transform(const float* __restrict__ pose_obs,
                                               const float* __restrict__ poses_last,
                                               float& ct, float& snt,
                                               float& tx, float& ty) {
  float sx, sy, nt;
  eval_pose(pose_obs, poses_last, sx, sy, nt);
  // st_xy = -(st*100/RES - half)/half, half = 240, RES = 5
  tx = -(sx * 20.0f - 240.0f) * (1.0f / 240.0f);
  ty = -(sy * 20.0f - 240.0f) * (1.0f / 240.0f);
  float t = (90.0f - nt) * RAD_PER_DEG;
  ct = cosf(t);
  snt = sinf(t);
}

// ---------------- kernel 1: zero the global accumulators ----------------
__global__ void init_kernel(float* __restrict__ ws) {
  int i = blockIdx.x * blockDim.x + threadIdx.x;
  if (i < 2 * VR * VR) ws[i] = 0.0f;
}

// ---------------- kernel 2: fused depth->points->trilinear splat ----------------
// Depth is streamed via double-buffered GLOBAL_LOAD_ASYNC_TO_LDS_B128 (ASYNCcnt),
// splat accumulates into LDS-private planes (ds_add_f32), flushed with
// global_atomic_add_f32.
__global__ void splat_kernel(const float* __restrict__ obs,
                             const float* __restrict__ agent_heights,
                             float* __restrict__ map_sum,
                             float* __restrict__ exp_sum) {
  extern __shared__ float lds[];
  float* lmap = lds;                         // 10000 floats
  float* lexp = lds + VR * VR;               // 10000 floats
  float* dbuf = lds + 2 * VR * VR;           // 2 x 1024 floats (double buffer)

  const int tid = threadIdx.x;
  for (int j = tid; j < VR * VR; j += SPLAT_THREADS) { lmap[j] = 0.0f; lexp[j] = 0.0f; }
  __syncthreads();

  const float F    = 640.0f / tanf(39.5f * RAD_PER_DEG);  // focal length
  const float invF = 1.0f / F;
  const float ah   = agent_heights[0];
  const float* depth = obs + DEPTH_OFF;

  const uint32_t lds_slot0 = (uint32_t)(uintptr_t)&dbuf[tid * PIX_PER_THREAD];
  const uint32_t lds_slot1 = (uint32_t)(uintptr_t)&dbuf[CHUNK_PIX + tid * PIX_PER_THREAD];
  const int lane_off = blockIdx.x * CHUNK_PIX + tid * PIX_PER_THREAD;

  // prologue: chunk 0 -> buffer 0 (each lane async-copies its own 16B slot)
  async_load_b128(lds_slot0, depth + lane_off);

  for (int k = 0; k < N_CHUNKS; ++k) {
    const int pbase = lane_off + k * CHUNK_STRIDE;
    if (k + 1 < N_CHUNKS) {
      // issue chunk k+1 into the other buffer, then wait until <=1 outstanding:
      // in-order completion guarantees chunk k has landed in LDS.
      async_load_b128(((k + 1) & 1) ? lds_slot1 : lds_slot0,
                      depth + pbase + CHUNK_STRIDE);
      wait_async_le1();
    } else {
      wait_async_0();
    }

    const float* buf = (k & 1) ? (dbuf + CHUNK_PIX) : dbuf;
    v4f d4 = *(const v4f*)&buf[tid * PIX_PER_THREAD];

    // 4 consecutive pixels starting at a 4-aligned index: same image row
    const int h  = pbase / W_IMG;
    const int w0 = pbase - h * W_IMG;
    const float zrow = (float)(511 - h);

    #pragma unroll
    for (int u = 0; u < PIX_PER_THREAD; ++u) {
      float d = d4[u];
      // pos_x = X/5, pos_y = Y/5, pos_z = Z/5 + 16 (derived from reference)
      float px = ((float)(w0 + u - 640) * d * invF + 250.0f) * 0.2f;
      float py = d * 0.2f;
      float pz = (zrow * d * invF + ah) * 0.2f + 16.0f;

      float fx = floorf(px), fy = floorf(py), fz = floorf(pz);

      #pragma unroll
      for (int cx = 0; cx < 2; ++cx) {
        float qx = fx + (float)cx;
        if (qx <= 0.0f || qx >= (float)VR) continue;   // safe: p>0 && p<dim (strict)
        float wx = 1.0f - fabsf(px - qx);
        #pragma unroll
        for (int cy = 0; cy < 2; ++cy) {
          float qy = fy + (float)cy;
          if (qy <= 0.0f || qy >= (float)VR) continue;
          float wxy = wx * (1.0f - fabsf(py - qy));
          int cell = (int)qy * VR + (int)qx;           // fp maps are [y][x]
          #pragma unroll
          for (int cz = 0; cz < 2; ++cz) {
            float qz = fz + (float)cz;
            if (qz <= 0.0f || qz >= (float)HZ) continue;
            float wv = wxy * (1.0f - fabsf(pz - qz));
            int iz = (int)qz;
            atomicAdd(&lexp[cell], wv);                // ds_add_f32
            if (iz >= MINZ && iz < MAXZ) atomicAdd(&lmap[cell], wv);
          }
        }
      }
    }
  }

  __syncthreads();
  for (int j = tid; j < VR * VR; j += SPLAT_THREADS) {
    float m = lmap[j], e = lexp[j];
    if (m != 0.0f) atomicAdd(&map_sum[j], m);          // global_atomic_add_f32
    if (e != 0.0f) atomicAdd(&exp_sum[j], e);
  }
}

// ---------------- kernel 3: clip sums, emit fp_map_pred ----------------
__global__ void finalize_kernel(float* __restrict__ map_sum,
                                float* __restrict__ exp_sum,
                                float* __restrict__ out_fp) {
  int j = blockIdx.x * blockDim.x + threadIdx.x;
  if (j >= VR * VR) return;
  float m = fminf(fmaxf(map_sum[j], 0.0f), 1.0f);
  float e = fminf(fmaxf(exp_sum[j], 0.0f), 1.0f);
  map_sum[j] = m;   // clipped in place: becomes agent_view channel 0 window
  exp_sum[j] = e;   // channel 1 window
  out_fp[j] = m;
}

// ---------------- kernel 4: pose integration (6 output floats) ----------------
__global__ void pose_kernel(const float* __restrict__ pose_obs,
                            const float* __restrict__ poses_last,
                            float* __restrict__ out_pose) {
  if (threadIdx.x != 0) return;
  float x, y, nt;
  eval_pose(pose_obs, poses_last, x, y, nt);
  out_pose[0] = x; out_pose[1] = y; out_pose[2] = nt;
  out_pose[3] = x; out_pose[4] = y; out_pose[5] = nt;
}

// ---------------- kernel 5: rotation grid_sample ----------------
// agent_view is implicit: channel c nonzero only in [240,340)x[190,290),
// values from the clipped 100x100 maps in ws.
__device__ __forceinline__ float av_tap(const float* __restrict__ A, int xi, int yi) {
  if (xi < WIN_X0 || xi >= WIN_X0 + VR || yi < WIN_Y0 || yi >= WIN_Y0 + VR) return 0.0f;
  return A[(yi - WIN_Y0) * VR + (xi - WIN_X0)];
}

__global__ void rot_kernel(const float* __restrict__ map_clip,
                           const float* __restrict__ exp_clip,
                           const float* __restrict__ pose_obs,
                           const float* __restrict__ poses_last,
                           float* __restrict__ rotated) {
  int idx = blockIdx.x * blockDim.x + threadIdx.x;
  if (idx >= 2 * MC2) return;
  int c = idx / MC2;
  int rem = idx - c * MC2;
  int r = rem / MAPC;
  int j = rem - r * MAPC;

  float ct, snt, tx, ty;
  eval_transform(pose_obs, poses_last, ct, snt, tx, ty);
  (void)tx; (void)ty;

  float xb = ((float)j + 0.5f) * (2.0f / MAPC) - 1.0f;
  float yb = ((float)r + 0.5f) * (2.0f / MAPC) - 1.0f;
  float gx = ct * xb - snt * yb;
  float gy = snt * xb + ct * yb;
  float xs = (gx + 1.0f) * (MAPC * 0.5f) - 0.5f;
  float ys = (gy + 1.0f) * (MAPC * 0.5f) - 0.5f;

  int x0 = (int)floorf(xs), y0 = (int)floorf(ys);
  float wx = xs - (float)x0, wy = ys - (float)y0;

  const float* A = (c == 0) ? map_clip : exp_clip;
  float v00 = av_tap(A, x0,     y0);
  float v01 = av_tap(A, x0 + 1, y0);
  float v10 = av_tap(A, x0,     y0 + 1);
  float v11 = av_tap(A, x0 + 1, y0 + 1);
  rotated[idx] = v00 * (1.0f - wx) * (1.0f - wy) + v01 * wx * (1.0f - wy)
               + v10 * (1.0f - wx) * wy          + v11 * wx * wy;
}

// ---------------- kernel 6: translation grid_sample + max(maps_last) ----------------
__device__ __forceinline__ float rot_tap(const float* __restrict__ R, int xi, int yi) {
  if (xi < 0 || xi >= MAPC || yi < 0 || yi >= MAPC) return 0.0f;
  return R[yi * MAPC + xi];
}

__global__ void trn_kernel(const float* __restrict__ rotated,
                           const float* __restrict__ maps_last,
                           const float* __restrict__ pose_obs,
                           const float* __restrict__ poses_last,
                           float* __restrict__ out_map) {
  int idx = blockIdx.x * blockDim.x + threadIdx.x;
  if (idx >= 4 * MC2) return;
  int c = idx / MC2;

  if (c >= 2) {
    // translated channels 2,3 are exactly zero; maps_last >= 0
    out_map[idx] = maps_last[idx];
    return;
  }

  int rem = idx - c * MC2;
  int r = rem / MAPC;
  int j = rem - r * MAPC;

  float ct, snt, tx, ty;
  eval_transform(pose_obs, poses_last, ct, snt, tx, ty);
  (void)ct; (void)snt;

  float xb = ((float)j + 0.5f) * (2.0f / MAPC) - 1.0f + tx;
  float yb = ((float)r + 0.5f) * (2.0f / MAPC) - 1.0f + ty;
  float xs = (xb + 1.0f) * (MAPC * 0.5f) - 0.5f;
  float ys = (yb + 1.0f) * (MAPC * 0.5f) - 0.5f;

  int x0 = (int)floorf(xs), y0 = (int)floorf(ys);
  float wx = xs - (float)x0, wy = ys - (float)y0;

  const float* R = rotated + c * MC2;
  float v00 = rot_tap(R, x0,     y0);
  float v01 = rot_tap(R, x0 + 1, y0);
  float v10 = rot_tap(R, x0,     y0 + 1);
  float v11 = rot_tap(R, x0 + 1, y0 + 1);
  float v = v00 * (1.0f - wx) * (1.0f - wy) + v01 * wx * (1.0f - wy)
          + v10 * (1.0f - wx) * wy          + v11 * wx * wy;

  out_map[idx] = fmaxf(maps_last[idx], v);
}

// ---------------- launch ----------------
extern "C" void kernel_launch(void* const* d_in, const int* in_sizes, int n_in,
                              void* d_out, int out_size, void* d_ws, size_t ws_size,
                              hipStream_t stream) {
  const float* obs           = (const float*)d_in[0];
  const float* pose_obs      = (const float*)d_in[1];
  const float* maps_last     = (const float*)d_in[2];
  const float* poses_last    = (const float*)d_in[3];
  const float* agent_heights = (const float*)d_in[4];
  float* out = (float*)d_out;
  float* ws  = (float*)d_ws;

  // 1. zero accumulators (harness poisons ws; must re-zero every launch)
  init_kernel<<<(2 * VR * VR + 255) / 256, 256, 0, stream>>>(ws);

  // 2. splat: 128 blocks x 256 threads (8 waves/block, wave32)
  //    dynamic LDS: 80000B accumulators + 8192B async depth double-buffer
  splat_kernel<<<SPLAT_BLOCKS, SPLAT_THREADS,
                 (2 * VR * VR + 2 * CHUNK_PIX) * sizeof(float), stream>>>(
      obs, agent_heights, ws + WS_MAP, ws + WS_EXP);

  // 3. clip + fp_map_pred output
  finalize_kernel<<<(VR * VR + 255) / 256, 256, 0, stream>>>(
      ws + WS_MAP, ws + WS_EXP, out + OUT_FP);

  // 4. pose outputs
  pose_kernel<<<1, 32, 0, stream>>>(pose_obs, poses_last, out + OUT_POSE);

  // 5. rotation pass -> ws
  rot_kernel<<<(2 * MC2 + 255) / 256, 256, 0, stream>>>(
      ws + WS_MAP, ws + WS_EXP, pose_obs, poses_last, ws + WS_ROT);

  // 6. translation pass + max -> map_pred output
  trn_kernel<<<(4 * MC2 + 255) / 256, 256, 0, stream>>>(
      ws + WS_ROT, maps_last, pose_obs, poses_last, out + OUT_MAP);
}